// SGCN_3195455668266
// MI455X (gfx1250) — compile-verified
//
#include <hip/hip_runtime.h>
#include <hip/hip_bf16.h>

typedef __attribute__((ext_vector_type(2))) float v2f;
typedef __attribute__((ext_vector_type(8))) float v8f;

#define N_NODES 40000
#define N_EDGES 640000
#define F_DIM   128

// ---------------------------------------------------------------------------
// 1) degree init: deg[i] = 1.0 (self-loop weight)
// ---------------------------------------------------------------------------
__global__ void sgc_deg_init(float* __restrict__ deg, int n) {
    int i = blockIdx.x * blockDim.x + threadIdx.x;
    if (i < n) deg[i] = 1.0f;
}

// ---------------------------------------------------------------------------
// 2) degree accumulation: deg[col[e]] += w[e]
// ---------------------------------------------------------------------------
__global__ void sgc_deg_acc(const int* __restrict__ col,
                            const float* __restrict__ w,
                            float* __restrict__ deg, int e_cnt) {
    int e = blockIdx.x * blockDim.x + threadIdx.x;
    if (e < e_cnt) atomicAdd(&deg[col[e]], w[e]);
}

// ---------------------------------------------------------------------------
// 3) deg^{-1/2}
// ---------------------------------------------------------------------------
__global__ void sgc_dinvsqrt(const float* __restrict__ deg,
                             float* __restrict__ dis, int n) {
    int i = blockIdx.x * blockDim.x + threadIdx.x;
    if (i < n) {
        float d = deg[i];
        dis[i] = (d > 0.0f) ? rsqrtf(fmaxf(d, 1e-30f)) : 0.0f;
    }
}

// ---------------------------------------------------------------------------
// 4) per-edge norm: norm[e] = dis[row]*w*dis[col]
// ---------------------------------------------------------------------------
__global__ void sgc_norm(const int* __restrict__ row,
                         const int* __restrict__ col,
                         const float* __restrict__ w,
                         const float* __restrict__ dis,
                         float* __restrict__ norm, int e_cnt) {
    int e = blockIdx.x * blockDim.x + threadIdx.x;
    if (e < e_cnt) norm[e] = dis[row[e]] * w[e] * dis[col[e]];
}

// ---------------------------------------------------------------------------
// 5) self-loop init of a hop: y[i,:] = dis[i]^2 * x[i,:]
//    one thread per (node, float4 chunk): n*32 threads
// ---------------------------------------------------------------------------
__global__ void sgc_selfloop(const float* __restrict__ xin,
                             const float* __restrict__ dis,
                             float* __restrict__ yout, int n) {
    int idx = blockIdx.x * blockDim.x + threadIdx.x;
    int i = idx >> 5;
    int q = idx & 31;
    if (i >= n) return;
    float d = dis[i];
    float cf = d * d;
    float4 v = ((const float4*)(xin + (size_t)i * F_DIM))[q];
    float4 o = make_float4(cf * v.x, cf * v.y, cf * v.z, cf * v.w);
    ((float4*)(yout + (size_t)i * F_DIM))[q] = o;
}

// ---------------------------------------------------------------------------
// 6) edge scatter: y[col[e],:] += norm[e] * x[row[e],:]
//    one lane per (edge, float4 chunk): E*32 threads; float4 gather + 4 atomics
// ---------------------------------------------------------------------------
__global__ void sgc_scatter(const float* __restrict__ xin,
                            float* __restrict__ yout,
                            const int* __restrict__ row,
                            const int* __restrict__ col,
                            const float* __restrict__ norm, int e_cnt) {
    long long idx = (long long)blockIdx.x * blockDim.x + threadIdx.x;
    long long e = idx >> 5;
    int q = (int)(idx & 31);
    if (e >= e_cnt) return;
    int j = row[e];
    int i = col[e];
    float cf = norm[e];
    float4 v = ((const float4*)(xin + (size_t)j * F_DIM))[q];
    float* yb = yout + (size_t)i * F_DIM + q * 4;
    atomicAdd(yb + 0, cf * v.x);
    atomicAdd(yb + 1, cf * v.y);
    atomicAdd(yb + 2, cf * v.z);
    atomicAdd(yb + 3, cf * v.w);
}

// ---------------------------------------------------------------------------
// 7) final GEMM: out[N,128] = X[N,128] @ W^T + b   (W is [F_OUT, F_IN] row-major)
//    fp32 WMMA 16x16x4. Block = 256 threads = 8 waves; wave w owns the 16x16
//    output tile at cols [16w, 16w+16); block owns 16 rows.
//    A lane layout (32-bit A 16x4): lanes 0-15 hold K={k0,k0+1}, lanes 16-31
//    hold K={k0+2,k0+3}; M = lane%16. B mirrors with N = lane%16.
//    D layout: VGPR v -> M = v + 8*(lane/16), N = lane%16.
// ---------------------------------------------------------------------------
__global__ void sgc_gemm_wmma(const float* __restrict__ X,
                              const float* __restrict__ W,
                              const float* __restrict__ bias,
                              float* __restrict__ out, int n) {
    int wave = threadIdx.x >> 5;
    int lane = threadIdx.x & 31;
    int half = lane >> 4;    // 0: K pair {0,1};  1: K pair {2,3}
    int l16  = lane & 15;

    int rowBase = blockIdx.x * 16;
    int oBase   = wave * 16;
    if (rowBase >= n) return;

    const float* xrow = X + (size_t)(rowBase + l16) * F_DIM; // A: M = l16
    const float* wrow = W + (size_t)(oBase + l16) * F_DIM;   // B: N = l16 (W row o = out col)

    v8f c = {};
    #pragma unroll
    for (int k0 = 0; k0 < F_DIM; k0 += 4) {
        int ka = k0 + half * 2;                 // even -> 8B aligned
        v2f a = *(const v2f*)(xrow + ka);
        v2f b = *(const v2f*)(wrow + ka);
        c = __builtin_amdgcn_wmma_f32_16x16x4_f32(
                /*neg_a=*/false, a, /*neg_b=*/false, b,
                /*c_mod=*/(short)0, c, /*reuse_a=*/false, /*reuse_b=*/false);
    }

    float bv = bias[oBase + l16];
    #pragma unroll
    for (int v = 0; v < 8; ++v) {
        int r = rowBase + v + half * 8;
        out[(size_t)r * F_DIM + oBase + l16] = c[v] + bv;
    }
}

// ---------------------------------------------------------------------------
// launcher
// ---------------------------------------------------------------------------
extern "C" void kernel_launch(void* const* d_in, const int* in_sizes, int n_in,
                              void* d_out, int out_size, void* d_ws, size_t ws_size,
                              hipStream_t stream) {
    const float* x  = (const float*)d_in[0];            // [N, 128]
    const int*   ei = (const int*)d_in[1];              // [2, E] flat: row | col
    const float* ew = (const float*)d_in[2];            // [E]
    const float* W  = (const float*)d_in[3];            // [128, 128]
    const float* b  = (const float*)d_in[4];            // [128]
    float* out = (float*)d_out;                         // [N, 128]

    const int N = N_NODES;
    const int E = N_EDGES;
    const int* row = ei;          // source j
    const int* col = ei + E;      // target i

    // workspace layout (floats): deg[N] | dis[N] | norm[E] | ybuf[N*128]
    float* ws   = (float*)d_ws;
    float* deg  = ws;
    float* dis  = ws + N;
    float* norm = ws + 2 * N;
    float* ybuf = ws + 2 * N + E;

    const int TPB = 256;
    int gN   = (N + TPB - 1) / TPB;            // node-wise
    int gE   = (E + TPB - 1) / TPB;            // edge-wise
    int gN32 = (N * 32 + TPB - 1) / TPB;       // node x float4-chunk
    long long e32 = (long long)E * 32;
    int gE32 = (int)((e32 + TPB - 1) / TPB);   // edge x float4-chunk
    int gG   = N / 16;                         // 16-row GEMM blocks (40000 % 16 == 0)

    // gcn_norm
    sgc_deg_init<<<gN, TPB, 0, stream>>>(deg, N);
    sgc_deg_acc<<<gE, TPB, 0, stream>>>(col, ew, deg, E);
    sgc_dinvsqrt<<<gN, TPB, 0, stream>>>(deg, dis, N);
    sgc_norm<<<gE, TPB, 0, stream>>>(row, col, ew, dis, norm, E);

    // hop 1: x -> d_out
    sgc_selfloop<<<gN32, TPB, 0, stream>>>(x, dis, out, N);
    sgc_scatter<<<gE32, TPB, 0, stream>>>(x, out, row, col, norm, E);

    // hop 2: d_out -> ybuf
    sgc_selfloop<<<gN32, TPB, 0, stream>>>(out, dis, ybuf, N);
    sgc_scatter<<<gE32, TPB, 0, stream>>>(out, ybuf, row, col, norm, E);

    // final linear: out = ybuf @ W^T + b   (fp32 WMMA)
    sgc_gemm_wmma<<<gG, TPB, 0, stream>>>(ybuf, W, b, out, N);
}